// TopKRouter_81750407512546
// MI455X (gfx1250) — compile-verified
//
#include <hip/hip_runtime.h>
#include <hip/hip_bf16.h>

typedef __attribute__((ext_vector_type(2))) float v2f;
typedef __attribute__((ext_vector_type(8))) float v8f;
typedef __attribute__((ext_vector_type(4))) unsigned int u32x4;
typedef __attribute__((ext_vector_type(8))) int i32x8;
typedef __attribute__((ext_vector_type(4))) int i32x4;

#define T_TOK   16384
#define DM      2048
#define N_EXP   64
#define KCHUNK  128
#define NCHUNK  (DM / KCHUNK)                 // 16
#define ROWF    (KCHUNK + 4)                  // LDS row stride: +4 DWORD pad -> conflict-free banks
#define CHUNK_FLOATS (N_EXP * ROWF)           // 8448 floats
#define CHUNK_BYTES  (CHUNK_FLOATS * 4)       // 33792 B per buffer

// flat output layout (reference return order)
#define OFF_IDX    0           // [16384,2]
#define OFF_PROBS  32768       // [16384,2]
#define OFF_LOGITS 65536       // [16384,64]
#define OFF_LOSS   1114112     // scalar
#define OFF_COUNTS 1114113     // [64]

// ---------------------------------------------------------------------------
// 0) zero the 128-float accumulator workspace (64 counts + 64 prob sums)
// ---------------------------------------------------------------------------
__global__ void init_ws_kernel(float* __restrict__ ws) {
    ws[threadIdx.x] = 0.0f;
}

// ---------------------------------------------------------------------------
// TDM: DMA a 64-expert x KCHUNK-float tile of gate_w into LDS.
// D# per CDNA5 ISA ch.8: group0 = {flags, lds_addr, global_addr, type},
// group1 = {data_size + LDS padding, tensor 2048x64, tile 128x64, stride 2048}.
// pad_enable with interval=128 DWORDs (code 6), amount=4 DWORDs (code 3):
// each 128-float row lands at stride 132 floats -> bank = 4*row (mod 64),
// so a B64 read across 32 lanes touches all 64 LDS banks exactly once.
// ---------------------------------------------------------------------------
__device__ __forceinline__ void tdm_load_gw_chunk(const float* gw, int k0,
                                                  unsigned lds_byte_off) {
    unsigned long long ga = (unsigned long long)(const void*)(gw + k0);
    u32x4 g0;
    g0[0] = 1u;                                   // count=1, user load descriptor
    g0[1] = lds_byte_off;                         // lds_addr (static LDS base == 0)
    g0[2] = (unsigned)(ga & 0xffffffffull);       // global_addr[31:0]
    g0[3] = (unsigned)((ga >> 32) & 0x01ffffffull)// global_addr[56:32]
          | (2u << 30);                           // type = 2 ("image")
    i32x8 g1;
    g1[0] = (int)((2u << 16)                      // data_size = 4 B
          | (1u << 20)                            // pad_enable
          | (6u << 22)                            // pad_interval: every 128 DWORDs
          | (3u << 25));                          // pad_amount: 4 DWORDs
    g1[1] = (int)(((unsigned)DM & 0xffffu) << 16);        // tensor_dim0[15:0]
    g1[2] = (int)((((unsigned)DM >> 16) & 0xffffu)        // tensor_dim0[31:16]
          | (((unsigned)N_EXP & 0xffffu) << 16));         // tensor_dim1[15:0]
    g1[3] = (int)(((unsigned)KCHUNK & 0xffffu) << 16);    // tile_dim0 = 128
    g1[4] = (int)((unsigned)N_EXP & 0xffffu);             // tile_dim1 = 64, tile_dim2 = 0
    g1[5] = (int)(unsigned)DM;                            // tensor_dim0_stride = 2048
    g1[6] = 0;
    g1[7] = 0;
    i32x4 gz = {0, 0, 0, 0};
#if __clang_major__ >= 23
    i32x8 gz8 = {0, 0, 0, 0, 0, 0, 0, 0};
    __builtin_amdgcn_tensor_load_to_lds(g0, g1, gz, gz, gz8, 0);
#else
    __builtin_amdgcn_tensor_load_to_lds(g0, g1, gz, gz, 0);
#endif
}

// ---------------------------------------------------------------------------
// 1) gate GEMM: logits[T,64] = x[T,2048] @ gate_w[64,2048]^T via
//    V_WMMA_F32_16X16X4_F32. One wave owns a 16-token x 64-expert stripe
//    (4 C tiles). Block = 256 threads = 8 waves = 128 tokens.
//    gate_w is staged through LDS by the Tensor Data Mover, double-buffered,
//    so the VMEM path only streams x (the 134 MB / 5.8 us HBM floor).
//    A layout: lanes 0-15 row M=lane, K={k,k+1}; lanes 16-31 same rows K={k+2,k+3}.
//    B layout (symmetric): lane holds column N=lane&15, same K split.
// ---------------------------------------------------------------------------
__global__ __launch_bounds__(256, 2) void gate_gemm_wmma(
    const float* __restrict__ x, const float* __restrict__ gw,
    float* __restrict__ logits)
{
    __shared__ float bsm[2][CHUNK_FLOATS];       // 2 x 33 KB double buffer

    const int lane = threadIdx.x & 31;
    const int wave = threadIdx.x >> 5;
    const int l16  = lane & 15;
    const int hk   = (lane >> 4) << 1;           // 0 or 2: K sub-offset per half-wave
    const int t0   = blockIdx.x * 128 + wave * 16;

    const float* __restrict__ a_ptr = x + (size_t)(t0 + l16) * DM + hk;

    v8f c0 = {}, c1 = {}, c2 = {}, c3 = {};

    if (threadIdx.x < 32) {
        tdm_load_gw_chunk(gw, 0, 0u);
        __builtin_amdgcn_s_wait_tensorcnt(0);
    }
    __syncthreads();

    for (int c = 0; c < NCHUNK; ++c) {
        const int buf = c & 1;
        // The optimizer cannot see that the TDM writes bsm (descriptor carries a
        // raw LDS offset). Escape the pointer + clobber memory so the ds_loads
        // below are kept and ordered after the tensor-wait/barrier.
        asm volatile("" :: "v"(&bsm[0][0]) : "memory");

        if ((c + 1 < NCHUNK) && threadIdx.x < 32)
            tdm_load_gw_chunk(gw, (c + 1) * KCHUNK, (unsigned)((buf ^ 1) * CHUNK_BYTES));

        const float* __restrict__ ap = a_ptr + c * KCHUNK;
        const float* __restrict__ bp = &bsm[buf][l16 * ROWF + hk];

#pragma unroll 8
        for (int k = 0; k < KCHUNK; k += 4) {
            v2f a  = *(const v2f*)(ap + k);
            v2f b0 = *(const v2f*)(bp + k);
            v2f b1 = *(const v2f*)(bp + 16 * ROWF + k);
            v2f b2 = *(const v2f*)(bp + 32 * ROWF + k);
            v2f b3 = *(const v2f*)(bp + 48 * ROWF + k);
            c0 = __builtin_amdgcn_wmma_f32_16x16x4_f32(false, a, false, b0, (short)0, c0, false, false);
            c1 = __builtin_amdgcn_wmma_f32_16x16x4_f32(false, a, false, b1, (short)0, c1, false, false);
            c2 = __builtin_amdgcn_wmma_f32_16x16x4_f32(false, a, false, b2, (short)0, c2, false, false);
            c3 = __builtin_amdgcn_wmma_f32_16x16x4_f32(false, a, false, b3, (short)0, c3, false, false);
        }

        if (threadIdx.x < 32) __builtin_amdgcn_s_wait_tensorcnt(0);
        __syncthreads();
    }

    // C layout: VGPR r, lanes 0-15 -> (M=r, N=lane); lanes 16-31 -> (M=r+8, N=lane-16)
    const int trow = t0 + (lane >> 4) * 8;
    float* __restrict__ dst = logits + (size_t)trow * N_EXP + l16;
#pragma unroll
    for (int r = 0; r < 8; ++r) {
        dst[(size_t)r * N_EXP +  0] = c0[r];
        dst[(size_t)r * N_EXP + 16] = c1[r];
        dst[(size_t)r * N_EXP + 32] = c2[r];
        dst[(size_t)r * N_EXP + 48] = c3[r];
    }
}

// ---------------------------------------------------------------------------
// 2) per-token top-2 + softmax + balance-loss partials.
//    One wave per token (lane l holds experts l and l+32). Block = 8 tokens.
// ---------------------------------------------------------------------------
__global__ __launch_bounds__(256) void router_topk(
    const float* __restrict__ logits,
    float* __restrict__ out_idx, float* __restrict__ out_probs,
    float* __restrict__ ws_counts, float* __restrict__ ws_probsum)
{
    __shared__ float s_cnt[N_EXP];
    __shared__ float s_ps[N_EXP];
    const int tid = threadIdx.x;
    if (tid < N_EXP) { s_cnt[tid] = 0.0f; s_ps[tid] = 0.0f; }
    __syncthreads();

    const int lane = tid & 31;
    const int wave = tid >> 5;
    const int t = blockIdx.x * 8 + wave;

    const float va = logits[(size_t)t * N_EXP + lane];
    const float vb = logits[(size_t)t * N_EXP + 32 + lane];

    // local top-2 (low index wins ties, like jax.lax.top_k)
    float v0, v1; int e0, e1;
    if (va >= vb) { v0 = va; e0 = lane;      v1 = vb; e1 = lane + 32; }
    else          { v0 = vb; e0 = lane + 32; v1 = va; e1 = lane;      }

    // butterfly merge of sorted pairs across the wave
#pragma unroll
    for (int off = 16; off >= 1; off >>= 1) {
        float ov0 = __shfl_xor(v0, off, 32);
        int   oe0 = __shfl_xor(e0, off, 32);
        float ov1 = __shfl_xor(v1, off, 32);
        int   oe1 = __shfl_xor(e1, off, 32);
        if (ov0 > v0 || (ov0 == v0 && oe0 < e0)) {
            float nv1; int ne1;
            if (v0 > ov1 || (v0 == ov1 && e0 < oe1)) { nv1 = v0;  ne1 = e0; }
            else                                     { nv1 = ov1; ne1 = oe1; }
            v0 = ov0; e0 = oe0; v1 = nv1; e1 = ne1;
        } else if (ov0 > v1 || (ov0 == v1 && oe0 < e1)) {
            v1 = ov0; e1 = oe0;
        }
    }

    // full softmax over 64 experts; max == v0 (global top-1)
    const float m  = v0;
    const float ea = __expf(va - m);
    const float eb = __expf(vb - m);
    float s = ea + eb;
#pragma unroll
    for (int off = 16; off >= 1; off >>= 1) s += __shfl_xor(s, off, 32);
    const float inv_s = 1.0f / s;

    atomicAdd(&s_ps[lane],      ea * inv_s);
    atomicAdd(&s_ps[lane + 32], eb * inv_s);

    if (lane == 0) {
        atomicAdd(&s_cnt[e0], 1.0f);
        atomicAdd(&s_cnt[e1], 1.0f);
        out_idx[2 * t]     = (float)e0;
        out_idx[2 * t + 1] = (float)e1;
        // softmax over the two picked logits (v0 >= v1)
        const float d = __expf(v1 - v0);
        const float r = 1.0f / (1.0f + d);
        out_probs[2 * t]     = r;
        out_probs[2 * t + 1] = d * r;
    }
    __syncthreads();

    if (tid < N_EXP) {
        atomicAdd(&ws_counts[tid],  s_cnt[tid]);
        atomicAdd(&ws_probsum[tid], s_ps[tid]);
    }
}

// ---------------------------------------------------------------------------
// 3) finalize: balance loss scalar + expert_counts output. 64 threads.
// ---------------------------------------------------------------------------
__global__ void finalize_kernel(const float* __restrict__ ws_counts,
                                const float* __restrict__ ws_probsum,
                                float* __restrict__ out_loss,
                                float* __restrict__ out_counts)
{
    __shared__ float red[N_EXP];
    const int tid = threadIdx.x;
    const float c = ws_counts[tid];
    const float p = ws_probsum[tid];
    out_counts[tid] = c;
    red[tid] = (c / ((float)T_TOK * 2.0f)) * (p / (float)T_TOK);
#pragma unroll
    for (int off = 32; off >= 1; off >>= 1) {
        __syncthreads();
        if (tid < off) red[tid] += red[tid + off];
    }
    __syncthreads();
    if (tid == 0) out_loss[0] = red[0] * (float)N_EXP * 0.01f;
}

// ---------------------------------------------------------------------------
extern "C" void kernel_launch(void* const* d_in, const int* in_sizes, int n_in,
                              void* d_out, int out_size, void* d_ws, size_t ws_size,
                              hipStream_t stream) {
    const float* x  = (const float*)d_in[0];
    const float* gw = (const float*)d_in[1];
    float* out    = (float*)d_out;
    float* idx    = out + OFF_IDX;
    float* probs  = out + OFF_PROBS;
    float* logits = out + OFF_LOGITS;
    float* loss   = out + OFF_LOSS;
    float* counts = out + OFF_COUNTS;
    float* wsf    = (float*)d_ws;   // [0..63] counts, [64..127] prob sums

    init_ws_kernel<<<1, 128, 0, stream>>>(wsf);
    gate_gemm_wmma<<<T_TOK / 128, 256, 0, stream>>>(x, gw, logits);
    router_topk<<<T_TOK / 8, 256, 0, stream>>>(logits, idx, probs, wsf, wsf + 64);
    finalize_kernel<<<1, 64, 0, stream>>>(wsf, wsf + 64, loss, counts);
}